// GATLayer_81149112090690
// MI455X (gfx1250) — compile-verified
//
#include <hip/hip_runtime.h>
#include <hip/hip_bf16.h>

#define N_NODES 50000
#define N_EDGES 800000
#define IN_DIM  128
#define OUT_DIM 64
#define N_HEADS 4

typedef __attribute__((ext_vector_type(2))) float v2f;
typedef __attribute__((ext_vector_type(8))) float v8f;

// ---------------------------------------------------------------------------
// Kernel 1: wh = x @ fc_w^T + fc_b via V_WMMA_F32_16X16X4_F32.
// Grid: 3125 M-tiles (50000/16), block = 128 threads = 4 waves, wave w owns
// N-tile w (OUT_DIM = 4*16). K loop: 128 = 32 steps of K=4.
// A 16x4 f32 layout: lanes 0-15 -> M=lane, K={0,1}; lanes 16-31 -> M=lane-16, K={2,3}.
// B 4x16 f32 layout mirrors A with N in place of M.
// C/D: VGPR r -> M = r (+8 for upper half-wave), N = lane%16.
// ---------------------------------------------------------------------------
__global__ void gat_wh_wmma_kernel(const float* __restrict__ x,
                                   const float* __restrict__ fc_w,
                                   const float* __restrict__ fc_b,
                                   float* __restrict__ wh) {
  const int lane  = threadIdx.x & 31;
  const int wave  = threadIdx.x >> 5;      // N tile: 0..3
  const int mtile = blockIdx.x;            // 0..3124
  const int l16   = lane & 15;
  const int kb    = (lane >> 4) * 2;       // K sub-offset: 0 or 2

  const float* __restrict__ xrow = x    + (size_t)(mtile * 16 + l16) * IN_DIM + kb;
  const float* __restrict__ wrow = fc_w + (size_t)(wave * 16 + l16) * IN_DIM + kb;

  v8f acc = {};
#pragma unroll 8
  for (int k0 = 0; k0 < IN_DIM; k0 += 4) {
    v2f a; a.x = xrow[k0]; a.y = xrow[k0 + 1];
    v2f b; b.x = wrow[k0]; b.y = wrow[k0 + 1];
    acc = __builtin_amdgcn_wmma_f32_16x16x4_f32(
        /*neg_a=*/false, a, /*neg_b=*/false, b,
        /*c_mod=*/(short)0, acc, /*reuse_a=*/false, /*reuse_b=*/false);
  }

  const int n     = wave * 16 + l16;
  const int mbase = mtile * 16 + ((lane >> 4) ? 8 : 0);
  const float bias = fc_b[n];
#pragma unroll
  for (int r = 0; r < 8; ++r) {
    wh[(size_t)(mbase + r) * OUT_DIM + n] = acc[r] + bias;
  }
}

// ---------------------------------------------------------------------------
// Kernel 2: ps = x @ a_src^T, pd = x @ a_dst^T  (N_HEADS each). Thread/node.
// attn_w is (N_HEADS, 2*IN_DIM): first IN_DIM cols = a_src, rest = a_dst.
// ---------------------------------------------------------------------------
__global__ void gat_proj_kernel(const float* __restrict__ x,
                                const float* __restrict__ attn_w,
                                float* __restrict__ ps,
                                float* __restrict__ pd) {
  const int n = blockIdx.x * blockDim.x + threadIdx.x;
  if (n >= N_NODES) return;
  const float* __restrict__ xr = x + (size_t)n * IN_DIM;
  float as[N_HEADS] = {0.f, 0.f, 0.f, 0.f};
  float ad[N_HEADS] = {0.f, 0.f, 0.f, 0.f};
#pragma unroll 4
  for (int k = 0; k < IN_DIM; ++k) {
    const float xv = xr[k];
#pragma unroll
    for (int h = 0; h < N_HEADS; ++h) {
      as[h] = fmaf(xv, attn_w[h * (2 * IN_DIM) + k], as[h]);
      ad[h] = fmaf(xv, attn_w[h * (2 * IN_DIM) + IN_DIM + k], ad[h]);
    }
  }
#pragma unroll
  for (int h = 0; h < N_HEADS; ++h) {
    ps[n * N_HEADS + h] = as[h];
    pd[n * N_HEADS + h] = ad[h];
  }
}

// ---------------------------------------------------------------------------
// Kernel 3: zero-fill (denominators + output accumulator).
// ---------------------------------------------------------------------------
__global__ void gat_zero_kernel(float* __restrict__ p, int n) {
  const int i = blockIdx.x * blockDim.x + threadIdx.x;
  if (i < n) p[i] = 0.f;
}

// ---------------------------------------------------------------------------
// Kernel 4: per-edge attention scores + denominator scatter-add.
// s_fwd = lrelu(ps[src]+pd[dst]+b); s_rev = lrelu(ps[dst]+pd[src]+b)
// denom[src] += s_fwd ; denom[dst] += s_rev  (per head)
// ---------------------------------------------------------------------------
__global__ void gat_edge_score_kernel(const long long* __restrict__ ei,
                                      const float* __restrict__ ps,
                                      const float* __restrict__ pd,
                                      const float* __restrict__ attn_b,
                                      float* __restrict__ s_fwd,
                                      float* __restrict__ denom) {
  const int e = blockIdx.x * blockDim.x + threadIdx.x;
  if (e >= N_EDGES) return;
  const int s = (int)ei[2 * (size_t)e];
  const int d = (int)ei[2 * (size_t)e + 1];
#pragma unroll
  for (int h = 0; h < N_HEADS; ++h) {
    const float b  = attn_b[h];
    float vf = ps[s * N_HEADS + h] + pd[d * N_HEADS + h] + b;
    vf = (vf > 0.f) ? vf : 0.2f * vf;
    float vr = ps[d * N_HEADS + h] + pd[s * N_HEADS + h] + b;
    vr = (vr > 0.f) ? vr : 0.2f * vr;
    s_fwd[(size_t)e * N_HEADS + h] = vf;
    unsafeAtomicAdd(&denom[s * N_HEADS + h], vf);
    unsafeAtomicAdd(&denom[d * N_HEADS + h], vr);
  }
}

// ---------------------------------------------------------------------------
// Kernel 5: message aggregation. One wave32 per edge: lanes cover the 64
// output dims (2 per lane), 4 heads -> 8 fp32 hardware atomics per lane
// directly into d_out (L2-resident).
// ---------------------------------------------------------------------------
__global__ void gat_edge_agg_kernel(const long long* __restrict__ ei,
                                    const float* __restrict__ s_fwd,
                                    const float* __restrict__ denom,
                                    const float* __restrict__ wh,
                                    float* __restrict__ out) {
  const int gid  = blockIdx.x * blockDim.x + threadIdx.x;
  const int e    = gid >> 5;
  const int lane = gid & 31;
  if (e >= N_EDGES) return;
  const int s = (int)ei[2 * (size_t)e];
  const int d = (int)ei[2 * (size_t)e + 1];

  float alpha[N_HEADS];
#pragma unroll
  for (int h = 0; h < N_HEADS; ++h) {
    alpha[h] = s_fwd[(size_t)e * N_HEADS + h] / denom[s * N_HEADS + h];
  }

  const float* __restrict__ whd = wh + (size_t)d * OUT_DIM;
  const float w0 = whd[lane];
  const float w1 = whd[lane + 32];

  float* __restrict__ outs = out + (size_t)s * (N_HEADS * OUT_DIM);
#pragma unroll
  for (int h = 0; h < N_HEADS; ++h) {
    unsafeAtomicAdd(&outs[h * OUT_DIM + lane],      alpha[h] * w0);
    unsafeAtomicAdd(&outs[h * OUT_DIM + lane + 32], alpha[h] * w1);
  }
}

// ---------------------------------------------------------------------------
// Kernel 6: ELU epilogue in place on d_out.
// ---------------------------------------------------------------------------
__global__ void gat_elu_kernel(float* __restrict__ out, int n) {
  const int i = blockIdx.x * blockDim.x + threadIdx.x;
  if (i < n) {
    const float v = out[i];
    out[i] = (v > 0.f) ? v : (__expf(v) - 1.f);
  }
}

extern "C" void kernel_launch(void* const* d_in, const int* in_sizes, int n_in,
                              void* d_out, int out_size, void* d_ws, size_t ws_size,
                              hipStream_t stream) {
  const float*     x      = (const float*)d_in[0];
  const long long* ei     = (const long long*)d_in[1];  // int64 edge_index
  const float*     attn_w = (const float*)d_in[2];
  const float*     attn_b = (const float*)d_in[3];
  const float*     fc_w   = (const float*)d_in[4];
  const float*     fc_b   = (const float*)d_in[5];
  float*           out    = (float*)d_out;

  // Workspace layout (floats): wh | ps | pd | s_fwd | denom  (~28 MB total)
  float* wsf   = (float*)d_ws;
  float* wh    = wsf;                                   // 50000*64
  float* ps    = wh    + (size_t)N_NODES * OUT_DIM;     // 50000*4
  float* pd    = ps    + (size_t)N_NODES * N_HEADS;     // 50000*4
  float* s_fwd = pd    + (size_t)N_NODES * N_HEADS;     // 800000*4
  float* denom = s_fwd + (size_t)N_EDGES * N_HEADS;     // 50000*4

  // 1) wh GEMM (WMMA): 3125 M-tiles x 4 wave-N-tiles
  gat_wh_wmma_kernel<<<N_NODES / 16, 128, 0, stream>>>(x, fc_w, fc_b, wh);

  // 2) attention projections
  gat_proj_kernel<<<(N_NODES + 255) / 256, 256, 0, stream>>>(x, attn_w, ps, pd);

  // 3) zero denom and output accumulator
  gat_zero_kernel<<<(N_NODES * N_HEADS + 255) / 256, 256, 0, stream>>>(denom, N_NODES * N_HEADS);
  {
    const int total = N_NODES * N_HEADS * OUT_DIM;      // 12.8M
    gat_zero_kernel<<<(total + 255) / 256, 256, 0, stream>>>(out, total);
  }

  // 4) edge scores + denominators
  gat_edge_score_kernel<<<(N_EDGES + 255) / 256, 256, 0, stream>>>(
      ei, ps, pd, attn_b, s_fwd, denom);

  // 5) message aggregation: one wave per edge (8 waves / 256-thread block)
  {
    const long long waves = N_EDGES;
    const int blocks = (int)((waves * 32 + 255) / 256);
    gat_edge_agg_kernel<<<blocks, 256, 0, stream>>>(ei, s_fwd, denom, wh, out);
  }

  // 6) ELU epilogue
  {
    const int total = N_NODES * N_HEADS * OUT_DIM;
    gat_elu_kernel<<<(total + 255) / 256, 256, 0, stream>>>(out, total);
  }
}